// PoseNDF_25898652795028
// MI455X (gfx1250) — compile-verified
//
#include <hip/hip_runtime.h>
#include <hip/hip_bf16.h>

typedef __attribute__((ext_vector_type(2))) float v2f;
typedef __attribute__((ext_vector_type(8))) float v8f;

#define NQ    1024          // query poses
#define MB    10000         // bank poses
#define NJ    21            // joints
#define QD    4             // quat dims
#define ROWF  (NJ * QD)     // 84 floats per pose row
#define KNEI  5
#define CLIPV 0.9999999f

// ---------------------------------------------------------------------------
// Phase 0: normalize each joint quaternion of `pose` into workspace.
// Flat layout of [N,21,4] and [N,84] is identical, so index by joint.
// ---------------------------------------------------------------------------
__global__ __launch_bounds__(256)
void pose_normalize_kernel(const float* __restrict__ pose,
                           float* __restrict__ rand_out, int njoints_total) {
    int idx = blockIdx.x * 256 + threadIdx.x;     // joint index: n*21 + j
    if (idx >= njoints_total) return;
    const float4 q = reinterpret_cast<const float4*>(pose)[idx];
    float d = q.x * q.x + q.y * q.y + q.z * q.z + q.w * q.w;
    float inv = __builtin_amdgcn_rsqf(d);         // raw v_rsq_f32
    float4 o;
    o.x = q.x * inv; o.y = q.y * inv; o.z = q.z * inv; o.w = q.w * inv;
    reinterpret_cast<float4*>(rand_out)[idx] = o;
}

// Branch-free acos (Abramowitz & Stegun 4.4.45): raw v_sqrt_f32 + 3 FMA + select.
// Clamp guarantees 1-|x| >= ~6e-8 (normal float), so the unscaled hardware
// sqrt is safe -- avoids the libm denorm-rescue sequence.
__device__ __forceinline__ float acos_fast(float x) {
    float ax = __builtin_fabsf(x);
    float p  = __builtin_fmaf(ax, -0.0187292986f, 0.0742610186f);
    p        = __builtin_fmaf(p, ax, -0.2121143788f);
    p        = __builtin_fmaf(p, ax,  1.5707962513f);
    float r  = p * __builtin_amdgcn_sqrtf(1.0f - ax);   // raw v_sqrt_f32
    return (x < 0.0f) ? (3.14159265358979f - r) : r;
}

// ---------------------------------------------------------------------------
// Phase 1: distance matrix via V_WMMA_F32_16X16X4_F32.
// One wave computes one 16x16 tile of dist. Per joint j:
//   A (16x4)  : A[row][k]  = rand[n0+row][4j+k]
//   B (4x16)  : B[k][col]  = train[m0+col][4j+k]
// Lane mapping (ISA 16x4 f32 A layout): row/col = lane&15,
// k-base = (lane>>4)*2  -> one b64 load per lane per matrix.
// D (16x16 f32, 8 VGPRs): lanes 0-15 -> M=i, N=lane; lanes 16-31 -> M=8+i.
// ---------------------------------------------------------------------------
__global__ __launch_bounds__(256)
void pose_dist_wmma_kernel(const float* __restrict__ rand_q,   // [NQ][84]
                           const float* __restrict__ train,    // [MB][84]
                           float* __restrict__ dist) {         // [NQ][MB]
    const int lane  = threadIdx.x & 31;
    const int wave  = threadIdx.x >> 5;
    const int mtile = blockIdx.x * 8 + wave;                   // 625 m-tiles
    if (mtile >= (MB / 16)) return;                            // uniform per wave: EXEC stays all-1s
    const int n0 = blockIdx.y << 4;
    const int m0 = mtile << 4;

    const int r  = lane & 15;             // row (A) / col (B) within tile
    const int kb = (lane >> 4) << 1;      // 0 for lanes 0-15, 2 for lanes 16-31

    const float* Ap = rand_q + (size_t)(n0 + r) * ROWF + kb;
    const float* Bp = train  + (size_t)(m0 + r) * ROWF + kb;

    float acc[8];
#pragma unroll
    for (int i = 0; i < 8; ++i) acc[i] = 0.0f;

#pragma unroll
    for (int j = 0; j < NJ; ++j) {
        const v2f a = *reinterpret_cast<const v2f*>(Ap + j * QD);  // global_load_b64
        const v2f b = *reinterpret_cast<const v2f*>(Bp + j * QD);  // global_load_b64
        v8f c = {0.f, 0.f, 0.f, 0.f, 0.f, 0.f, 0.f, 0.f};
        // D = A x B + 0 : the per-joint quaternion dots for this 16x16 tile
        c = __builtin_amdgcn_wmma_f32_16x16x4_f32(
                /*neg_a=*/false, a, /*neg_b=*/false, b,
                /*c_mod=*/(short)0, c, /*reuse_a=*/false, /*reuse_b=*/false);
#pragma unroll
        for (int i = 0; i < 8; ++i) {
            // single v_med3_num_f32, no canonicalize prologue
            float x = __builtin_amdgcn_fmed3f(c[i], -CLIPV, CLIPV);
            acc[i] += acos_fast(x);
        }
    }

    // Store tile: row = i + 8*(lane>=16), col = lane&15
    const int rowbase = n0 + ((lane >> 4) << 3);
    float* outp = dist + (size_t)rowbase * MB + m0 + (lane & 15);
#pragma unroll
    for (int i = 0; i < 8; ++i) {
        outp[(size_t)i * MB] = acc[i] * 0.5f;
    }
}

// ---------------------------------------------------------------------------
// Phase 2: per-row smallest-5 mean. One block (256 threads) per row.
// ---------------------------------------------------------------------------
__device__ __forceinline__ void top5_insert(float (&t)[KNEI], float v) {
    if (v < t[KNEI - 1]) {
        t[KNEI - 1] = v;
#pragma unroll
        for (int k = KNEI - 1; k > 0; --k) {
            if (t[k] < t[k - 1]) { float tmp = t[k]; t[k] = t[k - 1]; t[k - 1] = tmp; }
        }
    }
}

__global__ __launch_bounds__(256)
void top5_mean_kernel(const float* __restrict__ dist, float* __restrict__ out) {
    __shared__ float sh[256 * KNEI];
    const int n = blockIdx.x;
    const float* row = dist + (size_t)n * MB;

    float t[KNEI];
#pragma unroll
    for (int i = 0; i < KNEI; ++i) t[i] = 3.4e38f;

    for (int m = threadIdx.x; m < MB; m += 256) {
        top5_insert(t, row[m]);
    }
#pragma unroll
    for (int i = 0; i < KNEI; ++i) sh[threadIdx.x * KNEI + i] = t[i];
    __syncthreads();

    // stage 1: 32 threads each merge 8 lists (40 values) -> lists 0..31
    if (threadIdx.x < 32) {
        float b[KNEI];
#pragma unroll
        for (int i = 0; i < KNEI; ++i) b[i] = 3.4e38f;
        for (int src = threadIdx.x * 8; src < threadIdx.x * 8 + 8; ++src)
            for (int i = 0; i < KNEI; ++i) top5_insert(b, sh[src * KNEI + i]);
        __syncthreads();
#pragma unroll
        for (int i = 0; i < KNEI; ++i) sh[threadIdx.x * KNEI + i] = b[i];
    } else {
        __syncthreads();
    }
    __syncthreads();

    if (threadIdx.x == 0) {
        float b[KNEI];
#pragma unroll
        for (int i = 0; i < KNEI; ++i) b[i] = 3.4e38f;
        for (int s = 0; s < 32 * KNEI; ++s) top5_insert(b, sh[s]);
        float sum = 0.0f;
#pragma unroll
        for (int i = 0; i < KNEI; ++i) sum += b[i];
        out[n] = sum * (1.0f / KNEI);
    }
}

// ---------------------------------------------------------------------------
// Launch
// ---------------------------------------------------------------------------
extern "C" void kernel_launch(void* const* d_in, const int* in_sizes, int n_in,
                              void* d_out, int out_size, void* d_ws, size_t ws_size,
                              hipStream_t stream) {
    const float* pose  = (const float*)d_in[0];   // [1024,21,4] f32
    const float* train = (const float*)d_in[1];   // [10000,21,4] f32 (pre-normalized)
    float* out = (float*)d_out;                   // [1024] f32

    // workspace: normalized pose (NQ*84 f32) then dist matrix (NQ*MB f32, ~40MB, L2-resident)
    float* rand_q = (float*)d_ws;
    float* dist   = rand_q + (size_t)NQ * ROWF;

    // phase 0: normalize
    {
        int total = NQ * NJ;
        pose_normalize_kernel<<<(total + 255) / 256, 256, 0, stream>>>(pose, rand_q, total);
    }
    // phase 1: WMMA distance tiles (64 n-tiles x 625 m-tiles; 8 waves/block)
    {
        dim3 grid((MB / 16 + 7) / 8, NQ / 16);
        pose_dist_wmma_kernel<<<grid, 256, 0, stream>>>(rand_q, train, dist);
    }
    // phase 2: top-5 mean per row
    {
        top5_mean_kernel<<<NQ, 256, 0, stream>>>(dist, out);
    }
}